// TestLayer_33019708571694
// MI455X (gfx1250) — compile-verified
//
#include <hip/hip_runtime.h>
#include <stdint.h>

#define L_LEN   16384
#define N_LAM   2048
#define B_BATCH 16

#define TILE_L  256   // l values per block (== blockDim.x)
#define NCHUNK  8     // n chunks
#define CHUNK   256   // lambdas per chunk (CHUNK * NCHUNK == N_LAM)

typedef __attribute__((ext_vector_type(4))) unsigned int u32x4;
typedef __attribute__((ext_vector_type(8))) unsigned int u32x8;

// ---------------------------------------------------------------------------
// Kernel 1: params[n] = { Re(lambda), Im(lambda)^2 }
//   lambda = (x0 + i x1) * (y0 + i y1)
// ---------------------------------------------------------------------------
__global__ void cauchy_prep(const float2* __restrict__ x,
                            const float2* __restrict__ y,
                            float2* __restrict__ params) {
    int n = blockIdx.x * blockDim.x + threadIdx.x;
    if (n < N_LAM) {
        float2 a = x[n];
        float2 b = y[n];
        float br = a.x * b.x - a.y * b.y;
        float bi = a.x * b.y + a.y * b.x;
        params[n] = make_float2(br, bi * bi);
    }
}

// ---------------------------------------------------------------------------
// Kernel 2: partial Cauchy sums.
//   block (bx, by): l-tile bx, n-chunk by.
//   Wave 0 TDM-loads the chunk's {br, bi^2} pairs (2 KB) into LDS, then each
//   lane accumulates 256 terms for its own omega_l and stores a partial sum:
//   zpart[by * L + l] = sum_{n in chunk} (w-br)/((w-br)^2 + bi^2)
// ---------------------------------------------------------------------------
__global__ void cauchy_main(const float2* __restrict__ params,
                            float* __restrict__ zpart) {
    __shared__ float2 sp[CHUNK];

    const int tid   = threadIdx.x;
    const int l     = blockIdx.x * TILE_L + tid;
    const int chunk = blockIdx.y;

    // --- Tensor Data Mover: global -> LDS, one DMA per block (wave 0 only) ---
    if (tid < 32) {
        uint64_t gaddr   = (uint64_t)(uintptr_t)(const void*)(params + chunk * CHUNK);
        uint32_t lds_adr = (uint32_t)(uintptr_t)(void*)sp;  // low 32 bits = LDS offset

        // D# group 0 (4 SGPRs): count=1 | lds_addr | global_addr[56:0] | type=2
        u32x4 g0;
        g0[0] = 1u;                                    // count=1, user descriptor
        g0[1] = lds_adr;                               // LDS byte address
        g0[2] = (uint32_t)gaddr;                       // global_addr[31:0]
        g0[3] = (uint32_t)((gaddr >> 32) & 0x01FFFFFFu) | 0x80000000u; // [56:32] | type=2

        // D# group 1 (8 SGPRs): 2D tile of 512 f32 (CHUNK float2 pairs)
        u32x8 g1;
        g1[0] = 0x00020000u;          // workgroup_mask=0, data_size=2 (4 bytes)
        g1[1] = (512u << 16);         // tensor_dim0[15:0] = 512
        g1[2] = (1u << 16);           // tensor_dim0[31:16]=0, tensor_dim1[15:0]=1
        g1[3] = (512u << 16);         // tensor_dim1[31:16]=0, tile_dim0 = 512
        g1[4] = 1u;                   // tile_dim1 = 1, tile_dim2 = 0
        g1[5] = 512u;                 // tensor_dim0_stride[31:0] = 512
        g1[6] = 0u;                   // stride0 hi, stride1 lo
        g1[7] = 0u;                   // stride1 hi

        asm volatile("tensor_load_to_lds %0, %1"
                     :
                     : "s"(g0), "s"(g1)
                     : "memory");
        __builtin_amdgcn_s_wait_tensorcnt(0);
    }
    __syncthreads();

    const float om = (float)l;
    float acc = 0.0f;

#pragma unroll 8
    for (int n = 0; n < CHUNK; ++n) {
        float2 p   = sp[n];
        float  num = om - p.x;                          // w - br
        float  den = __builtin_fmaf(num, num, p.y);     // (w-br)^2 + bi^2
        acc = __builtin_fmaf(num, __builtin_amdgcn_rcpf(den), acc);
    }

    zpart[chunk * L_LEN + l] = acc;                     // plain store: deterministic
}

// ---------------------------------------------------------------------------
// Kernel 3: out[b, l] = u[b, l] + sum_c zpart[c][l]   (float4 vectorized)
//   grid.x covers l in groups of 4, grid.y covers batch b.
// ---------------------------------------------------------------------------
__global__ void cauchy_finish(const float* __restrict__ u,
                              const float* __restrict__ zpart,
                              float* __restrict__ out) {
    int l4 = (blockIdx.x * blockDim.x + threadIdx.x) * 4;
    int b  = blockIdx.y;

    float4 s = make_float4(0.f, 0.f, 0.f, 0.f);
#pragma unroll
    for (int c = 0; c < NCHUNK; ++c) {
        float4 v = *(const float4*)(zpart + c * L_LEN + l4);
        s.x += v.x; s.y += v.y; s.z += v.z; s.w += v.w;
    }

    const float* ur = u + (size_t)b * L_LEN + l4;
    float4 uv = *(const float4*)ur;
    float4 o  = make_float4(uv.x + s.x, uv.y + s.y, uv.z + s.z, uv.w + s.w);
    *(float4*)(out + (size_t)b * L_LEN + l4) = o;
}

// ---------------------------------------------------------------------------
extern "C" void kernel_launch(void* const* d_in, const int* in_sizes, int n_in,
                              void* d_out, int out_size, void* d_ws, size_t ws_size,
                              hipStream_t stream) {
    const float*  u = (const float*)d_in[0];       // (B, L)
    const float2* x = (const float2*)d_in[1];      // (N, 2) -> float2
    const float2* y = (const float2*)d_in[2];      // (N, 2) -> float2
    float* out = (float*)d_out;                    // (B, L)

    // Workspace layout: zpart[NCHUNK * L] floats, then params[N] float2
    float*  zpart  = (float*)d_ws;
    float2* params = (float2*)(zpart + NCHUNK * L_LEN);

    cauchy_prep<<<dim3(N_LAM / 256), dim3(256), 0, stream>>>(x, y, params);

    dim3 gmain(L_LEN / TILE_L, NCHUNK);            // (64, 8) blocks, 8 waves each
    cauchy_main<<<gmain, dim3(TILE_L), 0, stream>>>(params, zpart);

    dim3 gfin(L_LEN / 4 / 128, B_BATCH);           // (32, 16) blocks
    cauchy_finish<<<gfin, dim3(128), 0, stream>>>(u, zpart, out);
}